// GraphSAGEEncoder_6674379178176
// MI455X (gfx1250) — compile-verified
//
#include <hip/hip_runtime.h>
#include <hip/hip_bf16.h>

// GraphSAGE 2-layer forward for MI455X (gfx1250, wave32).
//
// Roofline: GEMMs total ~13.1 GFLOP; scatter phase moves ~2.5 GB through
// L2-resident atomics (x=25.6MB, h=51.2MB both fit the 192MB L2). The scatter
// dominates, so GEMM precision is kept exact-fp32 via V_WMMA_F32_16X16X4_F32
// (16x16 f32 accumulators, K=4 per instruction).

typedef float v2f __attribute__((ext_vector_type(2)));
typedef float v8f __attribute__((ext_vector_type(8)));

// ---------------------------------------------------------------- utilities

__global__ void zero_kernel(float4* __restrict__ p, long n4) {
    long i = (long)blockIdx.x * blockDim.x + threadIdx.x;
    if (i < n4) p[i] = make_float4(0.f, 0.f, 0.f, 0.f);
}

__global__ void count_kernel(const int* __restrict__ dst, float* __restrict__ cnt, int nE) {
    int e = blockIdx.x * blockDim.x + threadIdx.x;
    if (e < nE) atomicAdd(&cnt[dst[e]], 1.0f);
}

// Edge-parallel scatter: thread = (edge, 4-float chunk). Reads of X[src] are
// coalesced float4; atomics land contiguously -> L2 atomic units.
template <int F>
__global__ void scatter_add_kernel(const float* __restrict__ X,
                                   const int* __restrict__ src,
                                   const int* __restrict__ dst,
                                   float* __restrict__ agg, int nE) {
    constexpr int F4 = F / 4;
    long tid = (long)blockIdx.x * blockDim.x + threadIdx.x;
    int e = (int)(tid / F4);
    if (e >= nE) return;
    int q = (int)(tid % F4);
    int s = src[e];
    int d = dst[e];
    float4 v = reinterpret_cast<const float4*>(X + (size_t)s * F)[q];
    float* base = agg + (size_t)d * F + q * 4;
    atomicAdd(base + 0, v.x);
    atomicAdd(base + 1, v.y);
    atomicAdd(base + 2, v.z);
    atomicAdd(base + 3, v.w);
}

template <int F>
__global__ void mean_kernel(float* __restrict__ agg, const float* __restrict__ cnt, int nN) {
    constexpr int F4 = F / 4;
    long tid = (long)blockIdx.x * blockDim.x + threadIdx.x;
    int i = (int)(tid / F4);
    if (i >= nN) return;
    int q = (int)(tid % F4);
    float c = cnt[i];
    c = c > 1.0f ? c : 1.0f;
    float4* p = reinterpret_cast<float4*>(agg + (size_t)i * F) + q;
    float4 v = *p;
    v.x /= c; v.y /= c; v.z /= c; v.w /= c;
    *p = v;
}

// ---------------------------------------------------------------- SAGE GEMM
//
// Out[i,:NC] = act( Aagg[i,:K] @ Wl + bias + Aself[i,:K] @ Wr )
//
// Block = 256 threads = 8 waves; each wave owns a 16-row stripe, the block
// covers 128 rows x all NC columns, so A-rows are read from HBM exactly once.
// Each pass stages the wave's 16xK A-tile into LDS (row stride K+4 floats to
// avoid ds_load_b64 bank conflicts), then runs K/4 V_WMMA_F32_16X16X4_F32 per
// 16-col output tile. Weights (<=256KB) are L2/WGP$-resident.
template <int K, int NC, bool RELU>
__launch_bounds__(256)
__global__ void sage_gemm_kernel(const float* __restrict__ Aagg,
                                 const float* __restrict__ Aself,
                                 const float* __restrict__ Wl,
                                 const float* __restrict__ Wr,
                                 const float* __restrict__ bias,
                                 float* __restrict__ Out, int nrows) {
    extern __shared__ float lds[];
    constexpr int LDK = K + 4;      // padded LDS row stride (floats)
    constexpr int NT  = NC / 16;    // 16-wide output column tiles

    const int wave = threadIdx.x >> 5;
    const int lane = threadIdx.x & 31;
    const int rowbase = blockIdx.x * 128 + wave * 16;
    float* As = lds + (size_t)wave * 16 * LDK;

    // fragment coordinates (ISA 7.12.2, 32-bit layouts)
    const int mrow  = lane & 15;         // A: row within tile
    const int khalf = (lane >> 4) * 2;   // A/B: K sub-pair selector
    const int ncol0 = lane & 15;         // B/C/D: column within tile

    v8f acc[NT];
#pragma unroll
    for (int t = 0; t < NT; ++t) acc[t] = v8f{};

#pragma unroll
    for (int pass = 0; pass < 2; ++pass) {
        const float* Asrc = pass ? Aself : Aagg;
        const float* W    = pass ? Wr : Wl;

        __syncthreads();   // reuse of LDS buffer between passes
        // stage 16 x K A-tile (coalesced float4 global loads, rows clamped so
        // WMMA can run with full EXEC on the tail block)
        for (int idx = lane; idx < 16 * (K / 4); idx += 32) {
            int r  = idx / (K / 4);
            int c4 = idx % (K / 4);
            int grow = rowbase + r;
            grow = grow < nrows ? grow : nrows - 1;
            float4 v = reinterpret_cast<const float4*>(Asrc + (size_t)grow * K)[c4];
            *reinterpret_cast<float4*>(As + r * LDK + c4 * 4) = v;
        }
        __syncthreads();

#pragma unroll
        for (int t = 0; t < NT; ++t) {
            const float* wcol = W + t * 16 + ncol0;
            const float* arow = As + mrow * LDK + khalf;
            v8f c = acc[t];
#pragma unroll 8
            for (int k0 = 0; k0 < K; k0 += 4) {
                v2f a = *reinterpret_cast<const v2f*>(arow + k0);
                v2f b;
                b.x = wcol[(k0 + khalf) * NC];
                b.y = wcol[(k0 + khalf + 1) * NC];
                // D = A(16x4,f32) * B(4x16,f32) + C(16x16,f32)
                c = __builtin_amdgcn_wmma_f32_16x16x4_f32(
                        /*neg_a=*/false, a, /*neg_b=*/false, b,
                        /*c_mod=*/(short)0, c, /*reuse_a=*/false, /*reuse_b=*/false);
            }
            acc[t] = c;
        }
    }

    // epilogue: bias (+ ReLU), predicated stores for the row tail
    const int mbase = (lane >> 4) * 8;
#pragma unroll
    for (int t = 0; t < NT; ++t) {
        int col = t * 16 + ncol0;
        float bv = bias[col];
#pragma unroll
        for (int i = 0; i < 8; ++i) {
            int row = rowbase + mbase + i;
            if (row < nrows) {
                float v = acc[t][i] + bv;
                if (RELU) v = fmaxf(v, 0.0f);
                Out[(size_t)row * NC + col] = v;
            }
        }
    }
}

// ---------------------------------------------------------------- launcher

extern "C" void kernel_launch(void* const* d_in, const int* in_sizes, int n_in,
                              void* d_out, int out_size, void* d_ws, size_t ws_size,
                              hipStream_t stream) {
    (void)n_in; (void)out_size; (void)ws_size;

    constexpr int FIN = 128, HID = 256, FOUT = 128;
    const float* x   = (const float*)d_in[0];
    const int*   ei  = (const int*)d_in[1];
    const float* W1l = (const float*)d_in[2];
    const float* b1  = (const float*)d_in[3];
    const float* W1r = (const float*)d_in[4];
    const float* W2l = (const float*)d_in[5];
    const float* b2  = (const float*)d_in[6];
    const float* W2r = (const float*)d_in[7];
    float* out = (float*)d_out;

    const int N = in_sizes[0] / FIN;   // 50000
    const int E = in_sizes[1] / 2;     // 800000
    const int* src = ei;
    const int* dst = ei + E;

    // workspace layout (floats): agg1[N*128] | agg2[N*256] | cnt[N] | h[N*256]
    float* ws   = (float*)d_ws;
    float* agg1 = ws;
    float* agg2 = agg1 + (size_t)N * FIN;
    float* cnt  = agg2 + (size_t)N * HID;
    float* h    = cnt + N;

    const int TPB = 256;

    // 1. zero agg1, agg2, cnt (contiguous prefix of ws)
    {
        long nfloat = (long)N * (FIN + HID + 1);
        long n4 = (nfloat + 3) / 4;   // may spill into h region; h is fully overwritten
        int grid = (int)((n4 + TPB - 1) / TPB);
        zero_kernel<<<grid, TPB, 0, stream>>>((float4*)ws, n4);
    }

    // 2. in-degree counts (shared by both layers)
    count_kernel<<<(E + TPB - 1) / TPB, TPB, 0, stream>>>(dst, cnt, E);

    // 3. layer-1 scatter-sum of x into agg1
    {
        long threads = (long)E * (FIN / 4);
        scatter_add_kernel<FIN><<<(int)((threads + TPB - 1) / TPB), TPB, 0, stream>>>(
            x, src, dst, agg1, E);
    }
    // 4. mean
    {
        long threads = (long)N * (FIN / 4);
        mean_kernel<FIN><<<(int)((threads + TPB - 1) / TPB), TPB, 0, stream>>>(agg1, cnt, N);
    }
    // 5. h = relu(mean1 @ W1l + b1 + x @ W1r)
    {
        int grid = (N + 127) / 128;
        size_t shmem = (size_t)8 * 16 * (128 + 4) * sizeof(float);  // 67,584 B
        sage_gemm_kernel<128, 256, true><<<grid, TPB, shmem, stream>>>(
            agg1, x, W1l, W1r, b1, h, N);
    }
    // 6. layer-2 scatter-sum of h into agg2
    {
        long threads = (long)E * (HID / 4);
        scatter_add_kernel<HID><<<(int)((threads + TPB - 1) / TPB), TPB, 0, stream>>>(
            h, src, dst, agg2, E);
    }
    // 7. mean
    {
        long threads = (long)N * (HID / 4);
        mean_kernel<HID><<<(int)((threads + TPB - 1) / TPB), TPB, 0, stream>>>(agg2, cnt, N);
    }
    // 8. out = mean2 @ W2l + b2 + h @ W2r
    {
        int grid = (N + 127) / 128;
        size_t shmem = (size_t)8 * 16 * (256 + 4) * sizeof(float);  // 133,120 B
        sage_gemm_kernel<256, 128, false><<<grid, TPB, shmem, stream>>>(
            agg2, h, W2l, W2r, b2, out, N);
    }
}